// GraphTransformerBlock_55877524521517
// MI455X (gfx1250) — compile-verified
//
#include <hip/hip_runtime.h>
#include <math.h>

#define B_   16
#define S_   512
#define D_   256
#define H_   8
#define FF_  1024
#define DK_  32
#define NTOK (B_*S_)
#define NKT  (S_/16)      // 32 key tiles
#define BST  132          // padded LDS row stride (floats) for bias tile

typedef float v2f __attribute__((ext_vector_type(2)));
typedef float v8f __attribute__((ext_vector_type(8)));

// Exact-fp32 matrix op: D = A(16x4) * B(4x16) + C, wave32.
__device__ __forceinline__ v8f wmma4(v2f a, v2f b, v8f c) {
    return __builtin_amdgcn_wmma_f32_16x16x4_f32(false, a, false, b, (short)0, c, false, false);
}

// Async global -> LDS 128-bit copy (per lane), tracked by ASYNCcnt.
__device__ __forceinline__ void async_load_b128(const void* g, void* l) {
    unsigned      laddr = (unsigned)(size_t)l;            // low 32 bits of LDS-aperture addr = LDS byte offset
    unsigned long long gaddr = (unsigned long long)(size_t)g;
    asm volatile("global_load_async_to_lds_b128 %0, %1, off"
                 :: "v"(laddr), "v"(gaddr) : "memory");
}
__device__ __forceinline__ void wait_async_le2() { asm volatile("s_wait_asynccnt 0x2" ::: "memory"); }
__device__ __forceinline__ void wait_async_le0() { asm volatile("s_wait_asynccnt 0x0" ::: "memory"); }

// ---------------------------------------------------------------------------
// Generic fp32 WMMA GEMM: out = epilogue(A[MxK] @ W[KxN] + bias [+ resid])
// Each wave computes a 16x64 tile (4 C fragments, A fragment reused 4x).
// mode 0: plain store [M,N] (+resid if non-null)
// mode 1: exact GELU then store [M,N]
// mode 2: scatter into QKV layout [B,H,S,DK] (N must be D_=256)
// ---------------------------------------------------------------------------
__global__ __launch_bounds__(256) void gemm_f32_wmma(
    const float* __restrict__ A, const float* __restrict__ W,
    const float* __restrict__ bias, const float* __restrict__ resid,
    float* __restrict__ out, int M, int N, int K, int mode)
{
    const int wid = blockIdx.x * (blockDim.x >> 5) + (threadIdx.x >> 5);
    const int Nt = N >> 6;            // 64-wide wave tiles
    const int Mt = M >> 4;
    if (wid >= Mt * Nt) return;
    const int mt = wid / Nt, nt = wid % Nt;
    const int lane = threadIdx.x & 31, lm = lane & 15, lg = lane >> 4;

    v8f acc[4];
    acc[0] = v8f{}; acc[1] = v8f{}; acc[2] = v8f{}; acc[3] = v8f{};

    const float* arow = A + (size_t)(mt * 16 + lm) * K;
    for (int k0 = 0; k0 < K; k0 += 4) {
        v2f a;
        a[0] = arow[k0 + 2 * lg];
        a[1] = arow[k0 + 2 * lg + 1];
        const float* wr0 = W + (size_t)(k0 + 2 * lg) * N + nt * 64 + lm;
        const float* wr1 = wr0 + N;
        #pragma unroll
        for (int j = 0; j < 4; ++j) {
            v2f b; b[0] = wr0[j * 16]; b[1] = wr1[j * 16];
            acc[j] = wmma4(a, b, acc[j]);
        }
    }

    #pragma unroll
    for (int j = 0; j < 4; ++j) {
        #pragma unroll
        for (int vv = 0; vv < 8; ++vv) {
            const int row = mt * 16 + vv + 8 * lg;
            const int col = nt * 64 + j * 16 + lm;
            float val = acc[j][vv] + bias[col];
            if (mode == 0) {
                if (resid) val += resid[(size_t)row * N + col];
                out[(size_t)row * N + col] = val;
            } else if (mode == 1) {
                val = 0.5f * val * (1.0f + erff(val * 0.70710678118654752f));
                out[(size_t)row * N + col] = val;
            } else { // QKV scatter: token row -> [B,H,S,DK]
                const int bb = row >> 9, ss = row & (S_ - 1);
                const int hh = col >> 5, dk = col & (DK_ - 1);
                out[(((size_t)(bb * H_ + hh)) * S_ + ss) * DK_ + dk] = val;
            }
        }
    }
}

// ---------------------------------------------------------------------------
// Flash attention with edge bias + key padding mask.
// Block = (b, q-tile of 16); 8 waves = 8 heads.
// edge_bias tiles (16q x 16k x 8h = 8 KB) are streamed via async global->LDS
// b128 copies, double-buffered to overlap HBM with score WMMAs + softmax.
// ---------------------------------------------------------------------------
__global__ __launch_bounds__(256) void attn_kernel(
    const float* __restrict__ Q, const float* __restrict__ Kb,
    const float* __restrict__ Vb, const float* __restrict__ ebias,
    const unsigned char* __restrict__ mask, float* __restrict__ out)
{
    __shared__ __align__(16) float bias_sh[2][16][BST];  // double-buffered bias tile
    __shared__ float plds[8][256];                        // per-wave 16x16 P staging
    const int h    = threadIdx.x >> 5;
    const int lane = threadIdx.x & 31, lm = lane & 15, lg = lane >> 4;
    const int bb   = blockIdx.x >> 5;       // S_/16 == 32 q-tiles
    const int qt   = blockIdx.x & 31;

    const size_t bh = (size_t)(bb * H_ + h) * S_;   // row base into [B,H,S,DK]

    // Preload Q A-fragments (K-dim = 32 -> 8 wmma steps of 4)
    v2f qa[8];
    {
        const float* qrow = Q + (bh + qt * 16 + lm) * DK_;
        #pragma unroll
        for (int k0 = 0; k0 < 8; ++k0) {
            qa[k0][0] = qrow[4 * k0 + 2 * lg];
            qa[k0][1] = qrow[4 * k0 + 2 * lg + 1];
        }
    }

    float mrun[8], lrun[8];
    #pragma unroll
    for (int i = 0; i < 8; ++i) { mrun[i] = -3.0e38f; lrun[i] = 0.0f; }
    v8f o0 = v8f{}, o1 = v8f{};
    const float scale = 0.17677669529663689f;  // 1/sqrt(32)

    // Wave h cooperatively copies bias rows q = 2h, 2h+1 of the tile:
    // each row is 16 keys x 8 heads = 128 floats = 512 B = 32 lanes x 16 B.
    auto issue_bias = [&](int kt, int buf) {
        #pragma unroll
        for (int r = 0; r < 2; ++r) {
            const int qr = 2 * h + r;
            const int qq = qt * 16 + qr;
            const float* g = ebias + (((size_t)(bb * S_ + qq)) * S_ + kt * 16) * H_ + lane * 4;
            async_load_b128(g, &bias_sh[buf][qr][lane * 4]);
        }
    };

    issue_bias(0, 0);

    for (int kt = 0; kt < NKT; ++kt) {
        const int cur = kt & 1;
        if (kt + 1 < NKT) { issue_bias(kt + 1, cur ^ 1); wait_async_le2(); }
        else              { wait_async_le0(); }
        __syncthreads();   // (A) bias[cur] landed in LDS for all waves

        // scores tile = Q(16x32) @ K^T(32x16)
        v8f s = v8f{};
        const float* krow = Kb + (bh + kt * 16 + lm) * DK_;
        #pragma unroll
        for (int k0 = 0; k0 < 8; ++k0) {
            v2f bf; bf[0] = krow[4 * k0 + 2 * lg]; bf[1] = krow[4 * k0 + 2 * lg + 1];
            s = wmma4(qa[k0], bf, s);
        }

        const int key  = kt * 16 + lm;
        const bool msk = mask[bb * S_ + key] != 0;

        float p[8];
        #pragma unroll
        for (int vv = 0; vv < 8; ++vv) {
            float sv = s[vv] * scale + bias_sh[cur][vv + 8 * lg][lm * 8 + h];
            if (msk) sv = -3.0e38f;
            // row max across 16-lane group
            float r = sv;
            #pragma unroll
            for (int off = 8; off; off >>= 1) r = fmaxf(r, __shfl_xor(r, off));
            const float mnew  = fmaxf(mrun[vv], r);
            const float alpha = __expf(mrun[vv] - mnew);
            const float pv    = __expf(sv - mnew);
            float rs = pv;
            #pragma unroll
            for (int off = 8; off; off >>= 1) rs += __shfl_xor(rs, off);
            lrun[vv] = lrun[vv] * alpha + rs;
            mrun[vv] = mnew;
            o0[vv] *= alpha; o1[vv] *= alpha;
            p[vv] = pv;
        }

        // C-layout -> A-layout through LDS (per-wave region)
        #pragma unroll
        for (int vv = 0; vv < 8; ++vv)
            plds[h][(vv + 8 * lg) * 16 + lm] = p[vv];
        __syncthreads();   // (B) all bias[cur] reads + plds writes done

        // O += P(16x16) @ V(16x32)
        const float* vrow = Vb + (bh + kt * 16) * DK_;
        #pragma unroll
        for (int k0 = 0; k0 < 4; ++k0) {
            v2f a;
            a[0] = plds[h][lm * 16 + 4 * k0 + 2 * lg];
            a[1] = plds[h][lm * 16 + 4 * k0 + 2 * lg + 1];
            v2f b0, b1;
            b0[0] = vrow[(4 * k0 + 2 * lg) * DK_ + lm];
            b0[1] = vrow[(4 * k0 + 2 * lg + 1) * DK_ + lm];
            b1[0] = vrow[(4 * k0 + 2 * lg) * DK_ + 16 + lm];
            b1[1] = vrow[(4 * k0 + 2 * lg + 1) * DK_ + 16 + lm];
            o0 = wmma4(a, b0, o0);
            o1 = wmma4(a, b1, o1);
        }
    }

    #pragma unroll
    for (int vv = 0; vv < 8; ++vv) {
        const float inv = 1.0f / lrun[vv];
        const int qq = qt * 16 + vv + 8 * lg;
        float* orow = out + ((size_t)(bb * S_ + qq)) * D_ + h * DK_;
        orow[lm]      = o0[vv] * inv;
        orow[16 + lm] = o1[vv] * inv;
    }
}

// ---------------------------------------------------------------------------
// In-place LayerNorm over rows of length D_=256; one wave per row.
// ---------------------------------------------------------------------------
__global__ __launch_bounds__(256) void ln_kernel(
    float* __restrict__ x, const float* __restrict__ g,
    const float* __restrict__ bta, int M)
{
    const int row = blockIdx.x * (blockDim.x >> 5) + (threadIdx.x >> 5);
    if (row >= M) return;
    const int lane = threadIdx.x & 31;
    float* p = x + (size_t)row * D_;
    float vals[8];
    float sum = 0.0f;
    #pragma unroll
    for (int i = 0; i < 8; ++i) { vals[i] = p[i * 32 + lane]; sum += vals[i]; }
    #pragma unroll
    for (int off = 16; off; off >>= 1) sum += __shfl_xor(sum, off);
    const float mu = sum * (1.0f / D_);
    float vs = 0.0f;
    #pragma unroll
    for (int i = 0; i < 8; ++i) { const float d = vals[i] - mu; vs += d * d; }
    #pragma unroll
    for (int off = 16; off; off >>= 1) vs += __shfl_xor(vs, off);
    const float rs = rsqrtf(vs * (1.0f / D_) + 1e-5f);
    #pragma unroll
    for (int i = 0; i < 8; ++i) {
        const int c = i * 32 + lane;
        p[c] = (vals[i] - mu) * rs * g[c] + bta[c];
    }
}

// ---------------------------------------------------------------------------
extern "C" void kernel_launch(void* const* d_in, const int* in_sizes, int n_in,
                              void* d_out, int out_size, void* d_ws, size_t ws_size,
                              hipStream_t stream)
{
    const float* x    = (const float*)d_in[0];
    const float* eb   = (const float*)d_in[1];
    const unsigned char* mask = (const unsigned char*)d_in[2];
    const float* Wq = (const float*)d_in[3];  const float* bq  = (const float*)d_in[4];
    const float* Wk = (const float*)d_in[5];  const float* bk  = (const float*)d_in[6];
    const float* Wv = (const float*)d_in[7];  const float* bv  = (const float*)d_in[8];
    const float* Wo = (const float*)d_in[9];  const float* bo  = (const float*)d_in[10];
    const float* g1 = (const float*)d_in[11]; const float* b1  = (const float*)d_in[12];
    const float* Wf1= (const float*)d_in[13]; const float* bf1 = (const float*)d_in[14];
    const float* Wf2= (const float*)d_in[15]; const float* bf2 = (const float*)d_in[16];
    const float* g2 = (const float*)d_in[17]; const float* b2  = (const float*)d_in[18];
    float* out = (float*)d_out;

    float* q    = (float*)d_ws;
    float* kbuf = q    + (size_t)NTOK * D_;
    float* vbuf = kbuf + (size_t)NTOK * D_;
    float* attn = vbuf + (size_t)NTOK * D_;
    float* x1   = attn + (size_t)NTOK * D_;
    float* ff   = x1   + (size_t)NTOK * D_;   // NTOK*FF_ floats

    auto gemm = [&](const float* A, const float* W, const float* bias,
                    const float* resid, float* C, int M, int N, int K, int mode) {
        const int waves  = (M >> 4) * (N >> 6);
        const int blocks = (waves + 7) / 8;
        gemm_f32_wmma<<<blocks, 256, 0, stream>>>(A, W, bias, resid, C, M, N, K, mode);
    };

    // QKV projections into [B,H,S,DK]
    gemm(x, Wq, bq, nullptr, q,    NTOK, D_, D_, 2);
    gemm(x, Wk, bk, nullptr, kbuf, NTOK, D_, D_, 2);
    gemm(x, Wv, bv, nullptr, vbuf, NTOK, D_, D_, 2);

    // Flash attention with edge bias + padding mask
    attn_kernel<<<B_ * (S_ / 16), 256, 0, stream>>>(q, kbuf, vbuf, eb, mask, attn);

    // O-projection + residual, then LN1 (in place)
    gemm(attn, Wo, bo, x, x1, NTOK, D_, D_, 0);
    ln_kernel<<<NTOK / 8, 256, 0, stream>>>(x1, g1, b1, NTOK);

    // FFN: GELU(x1 @ Wf1 + bf1) @ Wf2 + bf2 + x1, then LN2 (in place on d_out)
    gemm(x1, Wf1, bf1, nullptr, ff, NTOK, FF_, D_, 1);
    gemm(ff, Wf2, bf2, x1, out, NTOK, D_, FF_, 0);
    ln_kernel<<<NTOK / 8, 256, 0, stream>>>(out, g2, b2, NTOK);
}